// CenternetDetections_51883204935755
// MI455X (gfx1250) — compile-verified
//
#include <hip/hip_runtime.h>
#include <cfloat>
#include <climits>
#include <cstdint>

// Problem constants (match reference)
#define BB      16
#define HH      128
#define WW      128
#define CC      80
#define KK      100          // MAX_OBJECTS
#define THRESH  0.1f         // SCORE_THRESHOLD
#define NBINS   4096
#define CAP     2048         // per-batch candidate capacity (~130 expected)
#define TILE    8
#define HALO    (TILE + 2)   // 10

// ---------------------------------------------------------------------------
// CDNA5 async global->LDS helpers (gfx1250 only)
// ---------------------------------------------------------------------------
__device__ __forceinline__ void cdna5_wait_async0() {
#if defined(__gfx1250__)
#if __has_builtin(__builtin_amdgcn_s_wait_asynccnt)
  __builtin_amdgcn_s_wait_asynccnt(0);
#else
  asm volatile("s_wait_asynccnt 0" ::: "memory");
#endif
#endif
}

// Per-lane async copy of 16 bytes from global memory to LDS.
// lds_addr = wave-relative LDS byte address (low 32 bits of a generic
// pointer to a __shared__ object ARE the LDS offset on CDNA5).
__device__ __forceinline__ void cdna5_async_load_b128(uint32_t lds_addr,
                                                      const void* gaddr) {
#if defined(__gfx1250__)
  asm volatile("global_load_async_to_lds_b128 %0, %1, off"
               :
               : "v"(lds_addr), "v"(gaddr)
               : "memory");
#endif
}

// ---------------------------------------------------------------------------
// Kernel 0: zero the histogram / counters region of the workspace
// ---------------------------------------------------------------------------
__global__ void CenternetZeroWs(int* __restrict__ p, int n) {
  int i = blockIdx.x * 256 + threadIdx.x;
  if (i < n) p[i] = 0;
}

// ---------------------------------------------------------------------------
// Kernel 1/3: NMS over an 8x8 spatial tile (all 80 channels) with a 1-pixel
// halo staged in LDS via async tensor-path copies.
//   phase==1 : histogram surviving maxima (> THRESH) into hist[b][bin]
//   phase==2 : append candidates with bin >= tbin[b] to the candidate buffer
// ---------------------------------------------------------------------------
__global__ __launch_bounds__(256)
void CenternetNmsPhase(const float* __restrict__ heat,
                       int*   __restrict__ hist,    // [BB*NBINS]
                       const int* __restrict__ tbin,// [BB]
                       int*   __restrict__ ccount,  // [BB]
                       float* __restrict__ cscore,  // [BB*CAP]
                       int*   __restrict__ cidx,    // [BB*CAP]
                       int phase) {
  __shared__ __align__(16) float tile[HALO * HALO * CC];  // 8000 f32 = 32 KB

  const int tid    = threadIdx.x;
  const int bb     = blockIdx.y;
  const int tileId = blockIdx.x;
  const int tX = tileId & (WW / TILE - 1);   // 0..15
  const int tY = tileId >> 4;                // 0..15
  const int ty0 = tY * TILE - 1;
  const int tx0 = tX * TILE - 1;

  // SAME padding: out-of-image halo acts like -inf for the max-pool.
  for (int i = tid; i < HALO * HALO * CC; i += 256) tile[i] = -FLT_MAX;
  __syncthreads();

  // Async-stage the valid part of the halo tile. For a fixed row, the x/c
  // extent is fully contiguous: (xhi-xlo)*80 floats, 320B per pixel,
  // 16B-aligned -> clean b128 segments.
  const int xlo = (tx0 < 0) ? 0 : tx0;
  const int xhi = (tx0 + HALO > WW) ? WW : (tx0 + HALO);
  const int segPerRow = (xhi - xlo) * (CC / 4);
  const uint32_t ldsBase = (uint32_t)(uintptr_t)&tile[0];

  for (int r = 0; r < HALO; ++r) {
    const int gy = ty0 + r;
    if (gy < 0 || gy >= HH) continue;
    const float* rowbase = heat + (((size_t)bb * HH + gy) * WW + xlo) * CC;
    const uint32_t ldsRow =
        ldsBase + (uint32_t)((r * HALO + (xlo - tx0)) * CC) * 4u;
    for (int s = tid; s < segPerRow; s += 256) {
      cdna5_async_load_b128(ldsRow + (uint32_t)s * 16u, rowbase + s * 4);
    }
  }
  cdna5_wait_async0();
  __syncthreads();

  const int myTb = (phase == 1) ? 0 : tbin[bb];

  // 8*8*80 = 5120 outputs, 20 per thread.
  for (int e = tid; e < TILE * TILE * CC; e += 256) {
    const int c  = e % CC;
    const int p  = e / CC;
    const int lx = p % TILE;
    const int ly = p / TILE;
    const int center = ((ly + 1) * HALO + (lx + 1)) * CC + c;

    const float v = tile[center];
    float m = v;
#pragma unroll
    for (int dy = -1; dy <= 1; ++dy) {
#pragma unroll
      for (int dx = -1; dx <= 1; ++dx) {
        m = fmaxf(m, tile[center + dy * HALO * CC + dx * CC]);
      }
    }
    // Local maximum survives NMS with its own score; only rows with
    // score > THRESH can appear non-zero in the reference output.
    if (v == m && v > THRESH) {
      int bin = (int)(v * (float)NBINS);
      bin = bin < 0 ? 0 : (bin >= NBINS ? NBINS - 1 : bin);
      if (phase == 1) {
        atomicAdd(&hist[bb * NBINS + bin], 1);
      } else if (bin >= myTb) {
        const int pos = atomicAdd(&ccount[bb], 1);
        if (pos < CAP) {
          const int gx  = tX * TILE + lx;
          const int gy2 = tY * TILE + ly;
          cscore[bb * CAP + pos] = v;
          cidx[bb * CAP + pos]   = (gy2 * WW + gx) * CC + c;
        }
      }
    }
  }
}

// ---------------------------------------------------------------------------
// Kernel 2: per-batch radix-select threshold: highest bin t such that the
// count of elements in bins >= t reaches KK (or 0 if fewer than KK total).
// ---------------------------------------------------------------------------
__global__ void CenternetThreshBin(const int* __restrict__ hist,
                                   int* __restrict__ tbin) {
  const int b = threadIdx.x;
  if (b < BB) {
    int cum = 0, t = 0;
    for (int bin = NBINS - 1; bin >= 0; --bin) {
      cum += hist[b * NBINS + bin];
      if (cum >= KK) { t = bin; break; }
    }
    tbin[b] = t;
  }
}

// ---------------------------------------------------------------------------
// Kernel 4: exact top-K over the small candidate set (ties broken by lower
// flat index, matching lax.top_k), decode boxes, write [KK,6] per batch.
// ---------------------------------------------------------------------------
__global__ __launch_bounds__(256)
void CenternetSelectDecode(const float* __restrict__ boxsz,
                           const float* __restrict__ offs,
                           const int*   __restrict__ ccount,
                           const float* __restrict__ cscore,
                           const int*   __restrict__ cidx,
                           float* __restrict__ out) {
  __shared__ float cs[CAP];
  __shared__ int   ci[CAP];
  __shared__ float rs[256];
  __shared__ int   ri[256];
  __shared__ int   rp[256];

  const int tid = threadIdx.x;
  const int bb  = blockIdx.x;
  int n = ccount[bb];
  if (n > CAP) n = CAP;

  for (int i = tid; i < n; i += 256) {
    cs[i] = cscore[bb * CAP + i];
    ci[i] = cidx[bb * CAP + i];
  }
  __syncthreads();

  for (int k = 0; k < KK; ++k) {
    // per-thread scan
    float bs = -1.0f;
    int bi = INT_MAX, bp = -1;
    for (int i = tid; i < n; i += 256) {
      const float s = cs[i];
      const int   id = ci[i];
      if (s > bs || (s == bs && id < bi)) { bs = s; bi = id; bp = i; }
    }
    rs[tid] = bs; ri[tid] = bi; rp[tid] = bp;
    __syncthreads();
    // tree reduce (score desc, index asc)
    for (int off = 128; off > 0; off >>= 1) {
      if (tid < off) {
        const float s2 = rs[tid + off];
        const int   i2 = ri[tid + off];
        if (s2 > rs[tid] || (s2 == rs[tid] && i2 < ri[tid])) {
          rs[tid] = s2; ri[tid] = i2; rp[tid] = rp[tid + off];
        }
      }
      __syncthreads();
    }

    if (tid == 0) {
      float* row = out + ((size_t)bb * KK + k) * 6;
      if (k < n) {
        const float sc  = rs[0];
        const int   idx = ri[0];
        cs[rp[0]] = -2.0f;  // consume
        const int c  = idx % CC;
        const int sp = idx / CC;
        const int xs = sp % WW;
        const int ys = sp / WW;
        const float* o = offs  + (((size_t)bb * HH * WW) + sp) * 2;
        const float* s2 = boxsz + (((size_t)bb * HH * WW) + sp) * 2;
        const float cy = (float)ys + o[0];
        const float cx = (float)xs + o[1];
        const float hb = s2[0] * 0.5f;
        const float wb = s2[1] * 0.5f;
        const float Hf = (float)HH, Wf = (float)WW;
        const float y1 = fminf(fmaxf(cy - hb, 0.0f), Hf) / Hf;
        const float y2 = fminf(fmaxf(cy + hb, 0.0f), Hf) / Hf;
        const float x1 = fminf(fmaxf(cx - wb, 0.0f), Wf) / Wf;
        const float x2 = fminf(fmaxf(cx + wb, 0.0f), Wf) / Wf;
        row[0] = y1; row[1] = x1; row[2] = y2; row[3] = x2;
        row[4] = sc; row[5] = (float)c;
      } else {
        // fewer than KK candidates above threshold -> zero rows
        row[0] = 0.f; row[1] = 0.f; row[2] = 0.f;
        row[3] = 0.f; row[4] = 0.f; row[5] = 0.f;
      }
    }
    __syncthreads();
  }
}

// ---------------------------------------------------------------------------
// Host launcher
// ---------------------------------------------------------------------------
extern "C" void kernel_launch(void* const* d_in, const int* in_sizes, int n_in,
                              void* d_out, int out_size, void* d_ws,
                              size_t ws_size, hipStream_t stream) {
  const float* heat  = (const float*)d_in[0];  // [16,128,128,80]
  const float* boxsz = (const float*)d_in[1];  // [16,128,128,2]
  const float* offs  = (const float*)d_in[2];  // [16,128,128,2]
  float* out = (float*)d_out;                  // [16,100,6]

  // Workspace layout (ints/floats, ~520 KB total)
  int*   hist   = (int*)d_ws;                 // BB*NBINS
  int*   tbin   = hist + BB * NBINS;          // BB
  int*   ccount = tbin + BB;                  // BB
  float* cscore = (float*)(ccount + BB);      // BB*CAP
  int*   cidxb  = (int*)(cscore + BB * CAP);  // BB*CAP

  const int nz = BB * NBINS + 2 * BB;
  CenternetZeroWs<<<(nz + 255) / 256, 256, 0, stream>>>(hist, nz);

  dim3 grid((WW / TILE) * (HH / TILE), BB);   // 256 tiles x 16 batches
  CenternetNmsPhase<<<grid, 256, 0, stream>>>(heat, hist, tbin, ccount,
                                              cscore, cidxb, /*phase=*/1);
  CenternetThreshBin<<<1, 32, 0, stream>>>(hist, tbin);
  CenternetNmsPhase<<<grid, 256, 0, stream>>>(heat, hist, tbin, ccount,
                                              cscore, cidxb, /*phase=*/2);
  CenternetSelectDecode<<<BB, 256, 0, stream>>>(boxsz, offs, ccount, cscore,
                                                cidxb, out);
}